// CollectDiffuseAttention_38233798869799
// MI455X (gfx1250) — compile-verified
//
#include <hip/hip_runtime.h>
#include <math.h>

typedef __attribute__((ext_vector_type(2))) float v2f;
typedef __attribute__((ext_vector_type(8))) float v8f;

constexpr int BATCH = 64;
constexpr int LEN   = 8192;
constexpr int DIM   = 64;

constexpr int NCHUNK        = 16;              // chunks along L
constexpr int CHUNK         = LEN / NCHUNK;    // 512 rows per block
constexpr int BLOCK         = 256;             // 8 waves (wave32)
constexpr int WAVES         = BLOCK / 32;
constexpr int ROWS_PER_WAVE = CHUNK / WAVES;   // 64
constexpr int GROUPS        = ROWS_PER_WAVE / 16;  // 4 x 16-row WMMA tiles

// workspace: ws1[B][NCHUNK][{m, s, pv[64]}] then ws2[B][64]
constexpr int    WS1_STRIDE = 2 + DIM;
constexpr size_t WS1_FLOATS = (size_t)BATCH * NCHUNK * WS1_STRIDE;

// ---------------------------------------------------------------------------
// Phase 1: logits = q.kc / tc (WMMA f32 16x16x4), streaming online softmax,
//          weighted-V partial sums per (batch, chunk).
// ---------------------------------------------------------------------------
__global__ __launch_bounds__(BLOCK) void collect_kernel(
    const float* __restrict__ q, const float* __restrict__ kc,
    const float* __restrict__ v, const float* __restrict__ tc,
    float* __restrict__ logits_out, float* __restrict__ ws1)
{
  __shared__ float q_s[DIM];
  __shared__ float redM[WAVES];
  __shared__ float redS[WAVES];
  __shared__ float pv_s[WAVES][DIM];

  const int b     = blockIdx.y;
  const int chunk = blockIdx.x;
  const int tid   = threadIdx.x;
  const int lane  = tid & 31;
  const int wave  = tid >> 5;

  if (tid < DIM) q_s[tid] = q[b * DIM + tid];
  __syncthreads();

  const float invTc = 1.0f / tc[0];

  // A fragment = q broadcast to all 16 M-rows.
  // f32 16x4 A layout: lanes 0-15 hold K=0(v0),K=1(v1); lanes 16-31 hold K=2,3.
  const int kSel = (lane >> 4) << 1;  // 0 or 2
  v2f afrag[16];
  #pragma unroll
  for (int k = 0; k < 16; ++k) {
    afrag[k].x = q_s[4 * k + kSel];
    afrag[k].y = q_s[4 * k + kSel + 1];
  }

  const int    laneRow = lane & 15;
  const size_t baseBL  = (size_t)b * LEN;

  float runM = -3.0e38f, runS = 0.0f;
  float pv0 = 0.0f, pv1 = 0.0f;

  for (int g = 0; g < GROUPS; ++g) {
    const int r0 = chunk * CHUNK + wave * ROWS_PER_WAVE + g * 16;

    // B fragment: 16x64 kc tile, N = row (lane&15), K split mirrors A.
    const float* rowPtr = kc + (baseBL + r0 + laneRow) * DIM + kSel;
    v8f c = {};
    #pragma unroll
    for (int k = 0; k < 16; ++k) {
      v2f bfrag = *(const v2f*)(rowPtr + 4 * k);
      c = __builtin_amdgcn_wmma_f32_16x16x4_f32(
              false, afrag[k], false, bfrag, (short)0, c, false, false);
    }
    // A-rows identical => every C element in lane t is logit[row t&15].
    const float logit = c[0] * invTc;

    if (lane < 16) logits_out[baseBL + r0 + laneRow] = logit;  // [B,L] output

    // Online softmax (16 distinct rows, duplicated across lane halves:
    // xor-reduce with offsets 8..1 stays within each 16-lane half).
    float gmax = logit;
    #pragma unroll
    for (int off = 8; off >= 1; off >>= 1)
      gmax = fmaxf(gmax, __shfl_xor(gmax, off, 32));

    const float newM  = fmaxf(runM, gmax);
    const float scale = __expf(runM - newM);
    const float w     = __expf(logit - newM);

    float sGroup = w;
    #pragma unroll
    for (int off = 8; off >= 1; off >>= 1)
      sGroup += __shfl_xor(sGroup, off, 32);

    runS = runS * scale + sGroup;
    pv0 *= scale;
    pv1 *= scale;
    runM = newM;

    // Weighted V: each lane owns columns 2*lane, 2*lane+1 (coalesced 256B rows)
    const float* vTile = v + (baseBL + r0) * DIM;
    #pragma unroll
    for (int r = 0; r < 16; ++r) {
      const float wr = __shfl(w, r, 32);
      v2f vv = *(const v2f*)(vTile + r * DIM + 2 * lane);
      pv0 = fmaf(wr, vv.x, pv0);
      pv1 = fmaf(wr, vv.y, pv1);
    }
  }

  if (lane == 0) { redM[wave] = runM; redS[wave] = runS; }
  pv_s[wave][2 * lane]     = pv0;
  pv_s[wave][2 * lane + 1] = pv1;
  __syncthreads();

  if (tid < DIM) {
    float M = redM[0];
    #pragma unroll
    for (int wv = 1; wv < WAVES; ++wv) M = fmaxf(M, redM[wv]);
    float S = 0.0f, acc = 0.0f;
    #pragma unroll
    for (int wv = 0; wv < WAVES; ++wv) {
      const float e = __expf(redM[wv] - M);
      S   += redS[wv] * e;
      acc += pv_s[wv][tid] * e;
    }
    float* dst = ws1 + ((size_t)b * NCHUNK + chunk) * WS1_STRIDE;
    if (tid == 0) dst[0] = M;
    if (tid == 1) dst[1] = S;   // S identical across threads
    dst[2 + tid] = acc;
  }
}

// ---------------------------------------------------------------------------
// Phase 2: merge chunk partials -> attn[b][64]
// ---------------------------------------------------------------------------
__global__ __launch_bounds__(64) void reduce_kernel(
    const float* __restrict__ ws1, float* __restrict__ attn)
{
  const int b = blockIdx.x;
  const int d = threadIdx.x;
  const float* base = ws1 + (size_t)b * NCHUNK * WS1_STRIDE;

  float M = -3.0e38f;
  #pragma unroll
  for (int c = 0; c < NCHUNK; ++c) M = fmaxf(M, base[c * WS1_STRIDE]);

  float S = 0.0f, acc = 0.0f;
  #pragma unroll
  for (int c = 0; c < NCHUNK; ++c) {
    const float e = __expf(base[c * WS1_STRIDE] - M);
    S   += base[c * WS1_STRIDE + 1] * e;
    acc += base[c * WS1_STRIDE + 2 + d] * e;
  }
  attn[b * DIM + d] = acc / S;
}

// ---------------------------------------------------------------------------
// Phase 3: gate = sigmoid(kd.q / td) via WMMA; out[b][l][:] = gate * attn[b][:]
// ---------------------------------------------------------------------------
__global__ __launch_bounds__(BLOCK) void diffuse_kernel(
    const float* __restrict__ q, const float* __restrict__ kd,
    const float* __restrict__ td, const float* __restrict__ attn,
    float* __restrict__ out)
{
  __shared__ float q_s[DIM];
  __shared__ float a_s[DIM];

  const int b     = blockIdx.y;
  const int chunk = blockIdx.x;
  const int tid   = threadIdx.x;
  const int lane  = tid & 31;
  const int wave  = tid >> 5;

  if (tid < DIM) {
    q_s[tid] = q[b * DIM + tid];
    a_s[tid] = attn[b * DIM + tid];
  }
  __syncthreads();

  const float invTd = 1.0f / td[0];

  const int kSel = (lane >> 4) << 1;
  v2f afrag[16];
  #pragma unroll
  for (int k = 0; k < 16; ++k) {
    afrag[k].x = q_s[4 * k + kSel];
    afrag[k].y = q_s[4 * k + kSel + 1];
  }

  const float  a0      = a_s[2 * lane];
  const float  a1      = a_s[2 * lane + 1];
  const int    laneRow = lane & 15;
  const size_t baseBL  = (size_t)b * LEN;

  for (int g = 0; g < GROUPS; ++g) {
    const int r0 = chunk * CHUNK + wave * ROWS_PER_WAVE + g * 16;

    const float* rowPtr = kd + (baseBL + r0 + laneRow) * DIM + kSel;
    v8f c = {};
    #pragma unroll
    for (int k = 0; k < 16; ++k) {
      v2f bfrag = *(const v2f*)(rowPtr + 4 * k);
      c = __builtin_amdgcn_wmma_f32_16x16x4_f32(
              false, afrag[k], false, bfrag, (short)0, c, false, false);
    }
    const float gl   = c[0] * invTd;
    const float gate = 1.0f / (1.0f + __expf(-gl));

    // 16 coalesced 256B row stores (the dominant 134MB write)
    float* oTile = out + (baseBL + r0) * DIM;
    #pragma unroll
    for (int r = 0; r < 16; ++r) {
      const float gr = __shfl(gate, r, 32);
      v2f ov;
      ov.x = gr * a0;
      ov.y = gr * a1;
      *(v2f*)(oTile + r * DIM + 2 * lane) = ov;
    }
  }
}

// ---------------------------------------------------------------------------
extern "C" void kernel_launch(void* const* d_in, const int* in_sizes, int n_in,
                              void* d_out, int out_size, void* d_ws, size_t ws_size,
                              hipStream_t stream)
{
  (void)in_sizes; (void)n_in; (void)out_size; (void)ws_size;

  const float* q  = (const float*)d_in[0];
  const float* kc = (const float*)d_in[1];
  const float* kd = (const float*)d_in[2];
  const float* v  = (const float*)d_in[3];
  const float* tc = (const float*)d_in[4];
  const float* td = (const float*)d_in[5];

  float* out    = (float*)d_out;                        // [B,L,D]
  float* logits = out + (size_t)BATCH * LEN * DIM;      // [B,L]

  float* ws1 = (float*)d_ws;
  float* ws2 = ws1 + WS1_FLOATS;

  const dim3 grid(NCHUNK, BATCH);
  collect_kernel<<<grid, BLOCK, 0, stream>>>(q, kc, v, tc, logits, ws1);
  reduce_kernel<<<dim3(BATCH), 64, 0, stream>>>(ws1, ws2);
  diffuse_kernel<<<grid, BLOCK, 0, stream>>>(q, kd, td, ws2, out);
}